// Consonance_78168404787675
// MI455X (gfx1250) — compile-verified
//
#include <hip/hip_runtime.h>

typedef float v2f __attribute__((ext_vector_type(2)));
typedef float v8f __attribute__((ext_vector_type(8)));

#define TPB   1024
#define NFFT  8192      // complex FFT length (even/odd packed real row of 16384)
#define NREAL 16384
#define KBINS 2048
#define MS    12
#define PI2   6.283185307179586f
// bank-conflict padding: +2 float2 every 32 (keeps b128 pairs contiguous & 16B aligned)
#define FP(a) ((a) + 2 * ((a) >> 5))
#define PADN  (NFFT + 2 * (NFFT / 32))   // 8704 float2 per buffer

// D = A(16x4) x B(4x16) + C, f32 matrix pipe
static __device__ __forceinline__ v8f wmma4(v2f a, v2f b, v8f c) {
  return __builtin_amdgcn_wmma_f32_16x16x4_f32(false, a, false, b, (short)0, c, false, false);
}

static __device__ __forceinline__ void cmul(float& xr, float& xi, float wr, float wi) {
  float r = xr * wr - xi * wi;
  float i = xr * wi + xi * wr;
  xr = r; xi = i;
}

// One Stockham radix-16 stage as complex 16x16 DFT-matrix GEMM on WMMA.
// SP = S' (sub-sequence count), L = current transform length, LP = 16*L.
// Read  addr: q' + SP*(r  + 16*k1) ; Write addr: q' + SP*(k2*L + k1)
// Twiddle exp(-2*pi*i*r*k1/LP) built as powers of T = exp(-2*pi*i*k1/LP):
// one sincos + complex-mult recurrence (T^2, T^4, running T^(4kk+2hi)).
template<int SP, int L, int LP>
static __device__ __forceinline__ void stage16(const float2* src, float2* dst,
                                               const v2f* aWr, const v2f* aWs, const v2f* aWn)
{
  const int tid  = (int)threadIdx.x;
  const int lane = tid & 31;
  const int wave = tid >> 5;          // 32 waves == 32 groups of 16 columns
  const int col  = lane & 15;
  const int hi   = lane >> 4;         // K-half / M-half selector per wmma layouts
  const int u    = (wave << 4) + col; // global column id in [0,512)
  const int qp   = u % SP;
  const int k1   = u / SP;
  const int base = qp + SP * 16 * k1;

  // T^1, T^2, T^4 and running power cur = T^(2hi)
  float s1, c1;
  __sincosf((-PI2 / (float)LP) * (float)k1, &s1, &c1);
  float c2 = c1 * c1 - s1 * s1, s2 = 2.f * c1 * s1;
  float c4 = c2 * c2 - s2 * s2, s4 = 2.f * c2 * s2;
  float ccur = hi ? c2 : 1.f;
  float scur = hi ? s2 : 0.f;

  v2f br[4], bi[4];
#pragma unroll
  for (int kk = 0; kk < 4; ++kk) {
    const int r0 = 4 * kk + 2 * hi;
    float2 z0 = src[FP(base + SP * r0)];
    float2 z1 = src[FP(base + SP * (r0 + 1))];
    // tw0 = T^(4kk+2hi) = cur ; tw1 = cur * T
    float ct1 = ccur, st1 = scur;
    cmul(ct1, st1, c1, s1);                 // (ct1,st1) = tw1
    float u0r = z0.x, u0i = z0.y;
    cmul(u0r, u0i, ccur, scur);
    float u1r = z1.x, u1i = z1.y;
    cmul(u1r, u1i, ct1, st1);
    br[kk][0] = u0r; br[kk][1] = u1r;
    bi[kk][0] = u0i; bi[kk][1] = u1i;
    cmul(ccur, scur, c4, s4);               // cur *= T^4
  }

  // Complex 16-pt DFT:  cr = cos*br + sin*bi ;  ci = -sin*br + cos*bi
  v8f cr = {}; v8f ci = {};
#pragma unroll
  for (int kk = 0; kk < 4; ++kk) cr = wmma4(aWr[kk], br[kk], cr);
#pragma unroll
  for (int kk = 0; kk < 4; ++kk) cr = wmma4(aWs[kk], bi[kk], cr);
#pragma unroll
  for (int kk = 0; kk < 4; ++kk) ci = wmma4(aWn[kk], br[kk], ci);
#pragma unroll
  for (int kk = 0; kk < 4; ++kk) ci = wmma4(aWr[kk], bi[kk], ci);

  // Scatter D (C/D layout: VGPR j -> M = j + 8*hi, N = col)
#pragma unroll
  for (int j = 0; j < 8; ++j) {
    const int k2 = j + 8 * hi;
    float2 o; o.x = cr[j]; o.y = ci[j];
    dst[FP(qp + SP * (k2 * L + k1))] = o;
  }
}

extern "C" __global__ void __launch_bounds__(TPB)
consonance_kernel(const float* __restrict__ Y,
                  const float* __restrict__ spec,
                  const float* __restrict__ diss,
                  float* __restrict__ out)
{
  extern __shared__ float2 lds2[];
  float2* Xc = lds2;          // [PADN]
  float2* Zc = lds2 + PADN;   // [PADN]

  const int tid  = (int)threadIdx.x;
  const int lane = tid & 31;
  const int row  = (int)blockIdx.x;

  // ---- async copy: global row -> LDS (even/odd packed complex), b128 chunks ----
  {
    const unsigned ldsbase = (unsigned)(uintptr_t)(void*)Xc;
    const unsigned long long grow = (unsigned long long)(Y + (size_t)row * NREAL);
#pragma unroll
    for (int ii = 0; ii < 4; ++ii) {
      const int p = 2 * (tid + TPB * ii);          // float2 pair index (b128 = 2 x float2)
      unsigned loff = ldsbase + 8u * (unsigned)FP(p);
      unsigned long long ga = grow + 8ull * (unsigned)p;
      asm volatile("global_load_async_to_lds_b128 %0, %1, off"
                   :: "v"(loff), "v"(ga) : "memory");
    }
    asm volatile("s_wait_asynccnt 0x0" ::: "memory");
  }

  // A operand: 16x16 DFT matrix W[k2][r] = exp(-2*pi*i*k2*r/16).
  // A layout: v[0] = A[m][4kk+2hi], v[1] = A[m][4kk+2hi+1], m = lane&15.
  v2f aWr[4], aWs[4], aWn[4];
  {
    const int m  = lane & 15;
    const int hi = lane >> 4;
#pragma unroll
    for (int kk = 0; kk < 4; ++kk) {
      const int r0 = 4 * kk + 2 * hi;
      float s0, c0, s1, c1;
      __sincosf((float)((m * r0) & 15) * (PI2 / 16.0f), &s0, &c0);
      __sincosf((float)((m * (r0 + 1)) & 15) * (PI2 / 16.0f), &s1, &c1);
      aWr[kk][0] = c0;  aWr[kk][1] = c1;   // Re(W)   = cos
      aWs[kk][0] = s0;  aWs[kk][1] = s1;   // -Im(W)  = +sin
      aWn[kk][0] = -s0; aWn[kk][1] = -s1;  //  Im(W)  = -sin
    }
  }
  __syncthreads();

  // Stage 1: radix-2, twiddle-free, in place (read/write same addresses)
#pragma unroll
  for (int i = 0; i < (NFFT / 2) / TPB; ++i) {
    const int q = tid + TPB * i;
    float2 a = Xc[FP(q)], b = Xc[FP(q + NFFT / 2)];
    float2 p, m2;
    p.x  = a.x + b.x;  p.y  = a.y + b.y;
    m2.x = a.x - b.x;  m2.y = a.y - b.y;
    Xc[FP(q)] = p; Xc[FP(q + NFFT / 2)] = m2;
  }
  __syncthreads();

  // Stages 2-4: radix-16 on the f32 WMMA pipe (Stockham ping-pong)
  stage16<256, 2, 32>(Xc, Zc, aWr, aWs, aWn);
  __syncthreads();
  stage16<16, 32, 512>(Zc, Xc, aWr, aWs, aWn);
  __syncthreads();
  stage16<1, 512, 8192>(Xc, Zc, aWr, aWs, aWn);
  __syncthreads();
  // Zc now holds Z = FFT_8192(even + i*odd), natural order.

  // Tail: unpack real-FFT, magnitudes, fused dot/norm accumulation.
  float s0 = 0.f, d[MS], s2[MS];
#pragma unroll
  for (int m = 0; m < MS; ++m) { d[m] = 0.f; s2[m] = 0.f; }

#pragma unroll
  for (int ii = 0; ii < KBINS / TPB; ++ii) {
    const int k = tid + TPB * ii;                  // k in [0,2048)
    float2 zk = Zc[FP(k)];
    float2 zm = Zc[FP((NFFT - k) & (NFFT - 1))];
    float Er = 0.5f * (zk.x + zm.x);
    float Ei = 0.5f * (zk.y - zm.y);
    float Or = 0.5f * (zk.y + zm.y);
    float Oi = 0.5f * (zm.x - zk.x);
    float s, c;
    __sincosf((-PI2 / (float)NREAL) * (float)k, &s, &c);
    float Xr = Er + c * Or - s * Oi;
    float Xi = Ei + c * Oi + s * Or;
    float mag = sqrtf(Xr * Xr + Xi * Xi) * (2.0f / (float)NREAL);
    s0 += mag * mag;
#pragma unroll
    for (int m = 0; m < MS; ++m) {
      float sv = spec[m * KBINS + k];              // L2-resident (98 KB total)
      d[m]  += mag * sv;
      s2[m] += sv * sv;
    }
  }

  // Block reduction of 25 partial sums (wave32 shfl, then LDS across waves)
  float vals[25];
  vals[0] = s0;
#pragma unroll
  for (int m = 0; m < MS; ++m) { vals[1 + m] = d[m]; vals[13 + m] = s2[m]; }
#pragma unroll
  for (int i = 0; i < 25; ++i) {
    float v = vals[i];
#pragma unroll
    for (int o = 16; o > 0; o >>= 1) v += __shfl_down(v, o, 32);
    vals[i] = v;
  }
  float* red = (float*)lds2;                       // Xc is dead; reuse as scratch
  const int wave = tid >> 5;
  if (lane == 0) {
#pragma unroll
    for (int i = 0; i < 25; ++i) red[wave * 25 + i] = vals[i];
  }
  __syncthreads();

  if (tid == 0) {
    float acc[25];
#pragma unroll
    for (int i = 0; i < 25; ++i) acc[i] = 0.f;
    for (int w = 0; w < TPB / 32; ++w) {
#pragma unroll
      for (int i = 0; i < 25; ++i) acc[i] += red[w * 25 + i];
    }
    float num = 0.f, den = 0.f;
#pragma unroll
    for (int m = 0; m < MS; ++m) {
      float sq   = acc[0] + acc[13 + m] - 2.0f * acc[1 + m];
      float dist = sqrtf(fmaxf(sq, 0.0f));
      float sim  = 1.0f / (dist + 1e-4f);
      num += sim * (1.0f - diss[m]);
      den += sim;
    }
    out[row] = num / den;
  }
}

extern "C" void kernel_launch(void* const* d_in, const int* in_sizes, int n_in,
                              void* d_out, int out_size, void* d_ws, size_t ws_size,
                              hipStream_t stream) {
  const float* Y    = (const float*)d_in[0];
  const float* spec = (const float*)d_in[1];
  const float* diss = (const float*)d_in[2];
  float* out = (float*)d_out;
  const int B = in_sizes[0] / NREAL;               // 4096 rows
  const size_t shmem = (size_t)2 * PADN * sizeof(float2);  // ~136 KB dynamic LDS
  consonance_kernel<<<dim3(B), dim3(TPB), shmem, stream>>>(Y, spec, diss, out);
}